// SelectiveSparseAttention_77764677861386
// MI455X (gfx1250) — compile-verified
//
#include <hip/hip_runtime.h>
#include <math.h>

// ---------------- problem constants ----------------
#define BB 4
#define LL 2048
#define DD 2048
#define HH 16
#define KVH 4
#define HD 128
#define KSEL 512
#define DH4 512            // D/4 (scorer hidden)
#define SCALE 0.08838834764831845f   // 128^-0.5
#define NEGV -1.0e9f

typedef __attribute__((ext_vector_type(2))) float v2f;
typedef __attribute__((ext_vector_type(8))) float v8f;
typedef __attribute__((ext_vector_type(4))) int   v4i;

static __device__ __forceinline__ v8f wmma4(v2f a, v2f b, v8f c) {
    // D(16x16,f32) = A(16x4,f32) x B(4x16,f32) + C
    return __builtin_amdgcn_wmma_f32_16x16x4_f32(false, a, false, b, (short)0, c, false, false);
}

// -------- CDNA5 async global->LDS staging (guarded; fallback = float4) -----
#if __has_builtin(__builtin_amdgcn_global_load_async_to_lds_b128) && \
    __has_builtin(__builtin_amdgcn_s_wait_asynccnt)
#define HAVE_ASYNC_LDS 1
#endif

typedef __attribute__((address_space(1))) v4i* gv4i_p;   // global v4i*
typedef __attribute__((address_space(3))) v4i* lv4i_p;   // LDS v4i*

static __device__ __forceinline__ void cp_b128(float* lds_dst, const float* gsrc) {
#ifdef HAVE_ASYNC_LDS
    __builtin_amdgcn_global_load_async_to_lds_b128(
        (gv4i_p)gsrc, (lv4i_p)lds_dst, 0, 0);
#else
    *(float4*)lds_dst = *(const float4*)gsrc;
#endif
}

static __device__ __forceinline__ void cp_wait() {
#ifdef HAVE_ASYNC_LDS
    __builtin_amdgcn_s_wait_asynccnt(0);
#endif
}

// =====================================================================
// fp32 WMMA GEMM: C(MxN) = act(A(MxK) @ B(KxN) + bias)
// block = 128 threads (4 waves), block tile = 64(M) x 64(N), K chunk = 32.
// Each wave owns a 64x16 column strip (4 accumulator tiles) so every
// staged B fragment is reused 4x from registers.
// rowmap (optional) indirects A rows (fused top-k gather).
// =====================================================================
#define TK 32
#define SA_STRIDE 36    // padded rows: conflict-free, 144B (16B-aligned)
#define SB_STRIDE 72    // padded rows: conflict-free, 288B (16B-aligned)

__global__ __launch_bounds__(128) void gemm_wmma_f32(
    const float* __restrict__ A, const float* __restrict__ Bm,
    float* __restrict__ C, int M, int N, int K,
    const float* __restrict__ bias, int act, const int* __restrict__ rowmap)
{
    __shared__ float sA[64 * SA_STRIDE];   // 9 KB
    __shared__ float sB[TK * SB_STRIDE];   // 9 KB

    const int tid   = threadIdx.x;
    const int wave  = tid >> 5;
    const int lane  = tid & 31;
    const int mlane = lane & 15;
    const int hi    = lane >> 4;
    const int klo   = hi * 2;
    const int m0    = blockIdx.y * 64;
    const int n0    = blockIdx.x * 64;

    v8f acc[4];
    #pragma unroll
    for (int mt = 0; mt < 4; ++mt) acc[mt] = (v8f){};

    for (int kk0 = 0; kk0 < K; kk0 += TK) {
        // stage A tile 64 x TK (512 float4, 4 per thread)
        for (int i = tid; i < (64 * TK) / 4; i += 128) {
            int lin = i * 4;
            int r = lin >> 5, c = lin & (TK - 1);
            int row = rowmap ? rowmap[m0 + r] : (m0 + r);
            cp_b128(&sA[r * SA_STRIDE + c], &A[(size_t)row * K + kk0 + c]);
        }
        // stage B tile TK x 64 (512 float4, 4 per thread)
        for (int i = tid; i < (TK * 64) / 4; i += 128) {
            int lin = i * 4;
            int r = lin >> 6, c = lin & 63;
            cp_b128(&sB[r * SB_STRIDE + c], &Bm[(size_t)(kk0 + r) * N + n0 + c]);
        }
        // prefetch next B chunk into L2 while we compute
        if (kk0 + TK < K)
            __builtin_prefetch(&Bm[(size_t)(kk0 + TK) * N + n0 + (tid & 63)], 0, 1);
        cp_wait();
        __syncthreads();

        #pragma unroll
        for (int s = 0; s < TK / 4; ++s) {
            const int kk = s * 4;
            v2f b;
            b.x = sB[(kk + klo) * SB_STRIDE + wave * 16 + mlane];
            b.y = sB[(kk + klo + 1) * SB_STRIDE + wave * 16 + mlane];
            #pragma unroll
            for (int mt = 0; mt < 4; ++mt) {
                v2f a;
                a.x = sA[(mt * 16 + mlane) * SA_STRIDE + kk + klo];
                a.y = sA[(mt * 16 + mlane) * SA_STRIDE + kk + klo + 1];
                acc[mt] = wmma4(a, b, acc[mt]);
            }
        }
        __syncthreads();
    }

    const int n = n0 + wave * 16 + mlane;
    const int mhi = hi * 8;
    const float bs = bias ? bias[n] : 0.0f;
    #pragma unroll
    for (int mt = 0; mt < 4; ++mt) {
        #pragma unroll
        for (int r = 0; r < 8; ++r) {
            float v = acc[mt][r] + bs;
            if (act == 1) v = 0.5f * v * (1.0f + erff(v * 0.70710678118654752f));
            C[(size_t)(m0 + mt * 16 + r + mhi) * N + n] = v;
        }
    }
}

// =====================================================================
// imp[tok] = h[tok,:] @ W2 + b2        (h: 8192 x 512)
// =====================================================================
__global__ __launch_bounds__(256) void imp_reduce_kernel(
    const float* __restrict__ h, const float* __restrict__ W2,
    const float* __restrict__ b2, float* __restrict__ imp)
{
    __shared__ float red[256];
    const int tok = blockIdx.x;
    const int t = threadIdx.x;
    float p = h[(size_t)tok * DH4 + t] * W2[t] +
              h[(size_t)tok * DH4 + t + 256] * W2[t + 256];
    red[t] = p;
    __syncthreads();
    for (int s = 128; s > 0; s >>= 1) {
        if (t < s) red[t] += red[t + s];
        __syncthreads();
    }
    if (t == 0) imp[tok] = red[0] + b2[0];
}

// =====================================================================
// Top-k (512 of 2048) per batch via bitonic sort; emits ascending idx,
// global row map for the gather-GEMM, and sigmoid gate.
// =====================================================================
__global__ __launch_bounds__(1024) void topk_kernel(
    const float* __restrict__ imp, int* __restrict__ idx_out,
    int* __restrict__ rowmap, float* __restrict__ gate)
{
    __shared__ float sval[LL];
    __shared__ int   sidx[LL];
    const int b = blockIdx.x;
    const int tid = threadIdx.x;

    for (int i = tid; i < LL; i += 1024) {
        sval[i] = imp[b * LL + i];
        sidx[i] = i;
    }
    __syncthreads();

    // sort (value desc, index asc tiebreak)
    for (int k = 2; k <= LL; k <<= 1) {
        for (int j = k >> 1; j > 0; j >>= 1) {
            for (int i = tid; i < LL; i += 1024) {
                int ixj = i ^ j;
                if (ixj > i) {
                    float v1 = sval[i], v2 = sval[ixj];
                    int   i1 = sidx[i], i2 = sidx[ixj];
                    bool g_i_first = (v1 > v2) || (v1 == v2 && i1 < i2);
                    bool up = ((i & k) == 0);          // descending block
                    bool doswap = up ? !g_i_first : g_i_first;
                    if (doswap) {
                        sval[i] = v2; sval[ixj] = v1;
                        sidx[i] = i2; sidx[ixj] = i1;
                    }
                }
            }
            __syncthreads();
        }
    }

    // re-sort top 512 indices ascending
    for (int k = 2; k <= KSEL; k <<= 1) {
        for (int j = k >> 1; j > 0; j >>= 1) {
            if (tid < KSEL) {
                int i = tid, ixj = i ^ j;
                if (ixj > i) {
                    int i1 = sidx[i], i2 = sidx[ixj];
                    bool up = ((i & k) == 0);          // ascending block
                    bool doswap = up ? (i1 > i2) : (i1 < i2);
                    if (doswap) { sidx[i] = i2; sidx[ixj] = i1; }
                }
            }
            __syncthreads();
        }
    }

    if (tid < KSEL) {
        int id = sidx[tid];
        idx_out[b * KSEL + tid] = id;
        rowmap [b * KSEL + tid] = b * LL + id;
        float v = imp[b * LL + id];
        gate[b * KSEL + tid] = 1.0f / (1.0f + expf(-v));
    }
}

// =====================================================================
// In-place RoPE. inv_freq(i) = theta^{-2i/HD} = exp(-ln(1e4)*i/64)
// =====================================================================
#define ROPE_C 0.14391156855135937f   // ln(10000)/64

__global__ __launch_bounds__(256) void rope_q_kernel(float* __restrict__ q)
{
    int gid = blockIdx.x * 256 + threadIdx.x;       // 8192*16*64 threads
    int i   = gid & 63;
    int h   = (gid >> 6) & 15;
    int row = gid >> 10;                            // 0..8191 (b*L + l)
    int pos = row & (LL - 1);
    float ang = (float)pos * __expf(-ROPE_C * (float)i);
    float c = cosf(ang), s = sinf(ang);
    float* p = q + (size_t)row * DD + h * HD + i;
    float x1 = p[0], x2 = p[64];
    p[0]  = x1 * c - x2 * s;
    p[64] = x1 * s + x2 * c;
}

__global__ __launch_bounds__(256) void rope_k_kernel(
    float* __restrict__ kv, const int* __restrict__ idx)
{
    int gid = blockIdx.x * 256 + threadIdx.x;       // 2048*4*64 threads
    int i   = gid & 63;
    int kvh = (gid >> 6) & 3;
    int row = gid >> 8;                             // 0..2047 (b*KSEL + t)
    int pos = idx[row];
    float ang = (float)pos * __expf(-ROPE_C * (float)i);
    float c = cosf(ang), s = sinf(ang);
    float* p = kv + (size_t)row * (2 * KVH * HD) + kvh * HD + i;  // K half
    float x1 = p[0], x2 = p[64];
    p[0]  = x1 * c - x2 * s;
    p[64] = x1 * s + x2 * c;
}

// =====================================================================
// Flash attention over the 512 gathered keys (GQA, causal on original
// key positions, importance gate). block = 128 thr (4 waves), each wave
// owns 16 query rows; grid = (L/64, H, B). Output in (B,L,D) layout.
// =====================================================================
#define KV_STRIDE 132     // padded 128-float rows (conflict free, 16B aligned)
#define P_STRIDE  17

__global__ __launch_bounds__(128) void attn_kernel(
    const float* __restrict__ q, const float* __restrict__ kv,
    const int* __restrict__ idx, const float* __restrict__ gate,
    float* __restrict__ out)
{
    __shared__ float sK[16 * KV_STRIDE];
    __shared__ float sV[16 * KV_STRIDE];
    __shared__ float sP[4 * 16 * P_STRIDE];

    const int b    = blockIdx.z;
    const int h    = blockIdx.y;
    const int kvh  = h >> 2;                 // GQA group
    const int tid  = threadIdx.x;
    const int wave = tid >> 5;
    const int lane = tid & 31;
    const int mlane = lane & 15;
    const int hi   = lane >> 4;
    const int klo  = hi * 2;
    const int mhi  = hi * 8;
    const int q0   = blockIdx.x * 64 + wave * 16;

    // preload this wave's 16x128 Q tile in WMMA A-layout registers
    v2f qa[32];
    const float* qrow = q + ((size_t)(b * LL + q0 + mlane)) * DD + h * HD;
    #pragma unroll
    for (int s = 0; s < 32; ++s) {
        qa[s].x = qrow[4 * s + klo];
        qa[s].y = qrow[4 * s + klo + 1];
    }

    v8f O[8];
    #pragma unroll
    for (int t = 0; t < 8; ++t) O[t] = (v8f){};
    float mrow[8], lrow[8];
    #pragma unroll
    for (int r = 0; r < 8; ++r) { mrow[r] = -1.0e30f; lrow[r] = 0.0f; }

    for (int t0 = 0; t0 < KSEL; t0 += 16) {
        // stage K/V 16x128 tiles via async global->LDS
        for (int i = tid; i < (16 * HD) / 4; i += 128) {
            int lin = i * 4;
            int r = lin >> 7, c = lin & 127;
            const float* src =
                kv + ((size_t)(b * KSEL + t0 + r)) * (2 * KVH * HD) + kvh * HD + c;
            cp_b128(&sK[r * KV_STRIDE + c], src);
            cp_b128(&sV[r * KV_STRIDE + c], src + KVH * HD);
        }
        cp_wait();
        __syncthreads();

        // scores: S(16x16) = Q(16x128) K^T(128x16)
        v8f sacc = {};
        #pragma unroll
        for (int s = 0; s < 32; ++s) {
            v2f bb;
            bb.x = sK[mlane * KV_STRIDE + 4 * s + klo];
            bb.y = sK[mlane * KV_STRIDE + 4 * s + klo + 1];
            sacc = wmma4(qa[s], bb, sacc);
        }

        const int   kp = idx [b * KSEL + t0 + mlane];   // this lane's key pos
        const float gg = gate[b * KSEL + t0 + mlane];

        #pragma unroll
        for (int r = 0; r < 8; ++r) {
            float v = sacc[r] * SCALE;
            int qpos = q0 + r + mhi;
            if (kp > qpos) v = NEGV;                    // causal on orig pos
            float mx = v;                               // row max (16-lane half)
            mx = fmaxf(mx, __shfl_xor(mx, 1));
            mx = fmaxf(mx, __shfl_xor(mx, 2));
            mx = fmaxf(mx, __shfl_xor(mx, 4));
            mx = fmaxf(mx, __shfl_xor(mx, 8));
            float mnew = fmaxf(mrow[r], mx);
            float p = __expf(v - mnew);
            float rs = p;
            rs += __shfl_xor(rs, 1);
            rs += __shfl_xor(rs, 2);
            rs += __shfl_xor(rs, 4);
            rs += __shfl_xor(rs, 8);
            float alpha = __expf(mrow[r] - mnew);
            lrow[r] = lrow[r] * alpha + rs;
            mrow[r] = mnew;
            #pragma unroll
            for (int nt = 0; nt < 8; ++nt) O[nt][r] *= alpha;
            // gated P into LDS (A-layout source for P@V)
            sP[wave * (16 * P_STRIDE) + (r + mhi) * P_STRIDE + mlane] = p * gg;
        }
        __syncthreads();

        // O += P(16x16) @ V(16x128)
        #pragma unroll
        for (int nt = 0; nt < 8; ++nt) {
            #pragma unroll
            for (int s4 = 0; s4 < 4; ++s4) {
                const int kk = 4 * s4;
                v2f a, bv;
                a.x = sP[wave * (16 * P_STRIDE) + mlane * P_STRIDE + kk + klo];
                a.y = sP[wave * (16 * P_STRIDE) + mlane * P_STRIDE + kk + klo + 1];
                bv.x = sV[(kk + klo) * KV_STRIDE + nt * 16 + mlane];
                bv.y = sV[(kk + klo + 1) * KV_STRIDE + nt * 16 + mlane];
                O[nt] = wmma4(a, bv, O[nt]);
            }
        }
        __syncthreads();
    }

    float invl[8];
    #pragma unroll
    for (int r = 0; r < 8; ++r) invl[r] = 1.0f / lrow[r];
    #pragma unroll
    for (int nt = 0; nt < 8; ++nt) {
        #pragma unroll
        for (int r = 0; r < 8; ++r) {
            out[((size_t)(b * LL + q0 + r + mhi)) * DD + h * HD + nt * 16 + mlane] =
                O[nt][r] * invl[r];
        }
    }
}

// =====================================================================
// launcher
// =====================================================================
extern "C" void kernel_launch(void* const* d_in, const int* in_sizes, int n_in,
                              void* d_out, int out_size, void* d_ws, size_t ws_size,
                              hipStream_t stream) {
    const float* x   = (const float*)d_in[0];
    const float* Wq  = (const float*)d_in[1];
    const float* Wkv = (const float*)d_in[2];
    const float* Wo  = (const float*)d_in[3];
    const float* W1  = (const float*)d_in[4];
    const float* b1  = (const float*)d_in[5];
    const float* W2  = (const float*)d_in[6];
    const float* b2  = (const float*)d_in[7];
    float* outp = (float*)d_out;

    // workspace carve-up (floats)
    float* ws    = (float*)d_ws;
    float* h     = ws;                          // 8192*512
    float* qraw  = h    + (size_t)8192 * 512;   // 8192*2048
    float* attn  = qraw + (size_t)8192 * 2048;  // 8192*2048
    float* kvraw = attn + (size_t)8192 * 2048;  // 2048*1024
    float* imp   = kvraw + (size_t)2048 * 1024; // 8192
    float* gate  = imp   + 8192;                // 2048
    int*   idx   = (int*)(gate + 2048);         // 2048
    int*   rmap  = idx + 2048;                  // 2048

    // 1. scorer hidden: h = gelu(x @ W1 + b1)
    gemm_wmma_f32<<<dim3(DH4 / 64, (BB * LL) / 64), 128, 0, stream>>>(
        x, W1, h, BB * LL, DH4, DD, b1, 1, nullptr);
    // 2. imp = h @ W2 + b2
    imp_reduce_kernel<<<BB * LL, 256, 0, stream>>>(h, W2, b2, imp);
    // 3. per-batch top-k -> idx / rowmap / gate
    topk_kernel<<<BB, 1024, 0, stream>>>(imp, idx, rmap, gate);
    // 4. kv = x[idx] @ Wkv   (gather fused via rowmap)
    gemm_wmma_f32<<<dim3((2 * KVH * HD) / 64, (BB * KSEL) / 64), 128, 0, stream>>>(
        x, Wkv, kvraw, BB * KSEL, 2 * KVH * HD, DD, nullptr, 0, rmap);
    // 5. q = x @ Wq
    gemm_wmma_f32<<<dim3(DD / 64, (BB * LL) / 64), 128, 0, stream>>>(
        x, Wq, qraw, BB * LL, DD, DD, nullptr, 0, nullptr);
    // 6. RoPE (in place)
    rope_q_kernel<<<(BB * LL * HH * 64) / 256, 256, 0, stream>>>(qraw);
    rope_k_kernel<<<(BB * KSEL * KVH * 64) / 256, 256, 0, stream>>>(kvraw, idx);
    // 7. flash attention over gathered keys
    attn_kernel<<<dim3(LL / 64, HH, BB), 128, 0, stream>>>(qraw, kvraw, idx, gate, attn);
    // 8. output projection
    gemm_wmma_f32<<<dim3(DD / 64, (BB * LL) / 64), 128, 0, stream>>>(
        attn, Wo, outp, BB * LL, DD, DD, nullptr, 0, nullptr);
}